// PPMIGCN_5789615915319
// MI455X (gfx1250) — compile-verified
//
#include <hip/hip_runtime.h>
#include <hip/hip_bf16.h>

typedef __attribute__((ext_vector_type(2))) float v2f;
typedef __attribute__((ext_vector_type(8))) float v8f;

#define NEG_SLOPE 0.2f

__device__ __forceinline__ unsigned fkey(float f) {
  unsigned u = __float_as_uint(f);
  return (u & 0x80000000u) ? ~u : (u | 0x80000000u);  // order-preserving
}
__device__ __forceinline__ float funkey(unsigned k) {
  unsigned u = (k & 0x80000000u) ? (k & 0x7fffffffu) : ~k;
  return __uint_as_float(u);
}
__device__ __forceinline__ float lrelu(float x) { return x > 0.0f ? x : NEG_SLOPE * x; }

// ---------------- GEMM: C[N,F] = A[N,K] @ W[K,F], fp32 WMMA 16x16x4 ----------------
// LDS holds W with K-row pairs interleaved per column:
//   lds2[(k/2)*SP + col] = { W[k][col], W[k+1][col] }   (float2, SP = F+16)
// so a B fragment is one aligned ds_load_b64 into an adjacent VGPR pair, and the
// pair-row stride (2F+32 dwords == 32 mod 64) makes the two lane-halves hit
// disjoint bank sets -> conflict-free.
template <int K, int F>
__global__ __launch_bounds__(256) void gemm_wmma(const float* __restrict__ A,
                                                 const float* __restrict__ W,
                                                 float* __restrict__ C, int Nn) {
  constexpr int SP = F + 16;         // pair-row stride in float2 units
  constexpr int NT = F / 16;
  __shared__ float w[(K / 2) * SP * 2];
  const int tid = threadIdx.x;
  for (int idx = tid; idx < K * F; idx += 256) {
    const int r = idx / F, c = idx - r * F;
    w[((r >> 1) * SP + c) * 2 + (r & 1)] = W[idx];
  }
  __syncthreads();

  const int wave = tid >> 5, lane = tid & 31;
  const int tile = blockIdx.x * 8 + wave;
  const int nTiles = (Nn + 15) >> 4;
  if (tile >= nTiles) return;        // wave-uniform: EXEC all-ones for WMMA

  const int half = lane >> 4, l15 = lane & 31 & 15;
  int row = tile * 16 + l15;
  if (row >= Nn) row = Nn - 1;       // clamp (tail tile); stores guarded below
  const float* arow = A + (size_t)row * K;
  const float2* w2 = (const float2*)w;

  v8f acc[NT] = {};
#pragma unroll 4
  for (int kk = 0; kk < K / 4; ++kk) {
    const int kb = kk * 4 + half * 2;          // even
    if ((kk & 7) == 0) __builtin_prefetch(arow + ((kb + 32) & (K - 1)), 0, 0);
    const float2 av = *(const float2*)(arow + kb);   // A frag: K = kb, kb+1
    v2f a; a.x = av.x; a.y = av.y;
    const float2* brow = w2 + (kb >> 1) * SP + l15;
#pragma unroll
    for (int nt = 0; nt < NT; ++nt) {
      const float2 bv = brow[nt * 16];               // {W[kb][col], W[kb+1][col]}
      v2f b; b.x = bv.x; b.y = bv.y;
      acc[nt] = __builtin_amdgcn_wmma_f32_16x16x4_f32(false, a, false, b,
                                                      (short)0, acc[nt], false, false);
    }
  }
#pragma unroll
  for (int nt = 0; nt < NT; ++nt)
#pragma unroll
    for (int r = 0; r < 8; ++r) {                    // D: VGPR r -> M = r + half*8
      const int orow = tile * 16 + half * 8 + r;
      if (orow < Nn) C[(size_t)orow * F + nt * 16 + l15] = acc[nt][r];
    }
}

// ---------------- per-node attention dots (wave per node) ----------------
__global__ __launch_bounds__(256) void dots_kernel(const float* __restrict__ xl,
                                                   const float* __restrict__ att_s,
                                                   const float* __restrict__ att_d,
                                                   float* __restrict__ a_src,
                                                   float* __restrict__ a_dst,
                                                   int N, int F) {
  const int i = blockIdx.x * 8 + (threadIdx.x >> 5);
  if (i >= N) return;
  const int lane = threadIdx.x & 31;
  const float* xr = xl + (size_t)i * F;
  float ps = 0.0f, pd = 0.0f;
  for (int h = lane; h < F; h += 32) {
    const float v = xr[h];
    ps += v * att_s[h];
    pd += v * att_d[h];
  }
  for (int m = 16; m >= 1; m >>= 1) {
    ps += __shfl_xor(ps, m, 32);
    pd += __shfl_xor(pd, m, 32);
  }
  if (lane == 0) { a_src[i] = ps; a_dst[i] = pd; }
}

// ---------------- scalar k_edge = dot(W_edge[0,:], att_edge) ----------------
__global__ void kedge_kernel(const float* __restrict__ We, const float* __restrict__ ae,
                             int F, float* __restrict__ out) {
  const int lane = threadIdx.x & 31;
  float p = 0.0f;
  for (int h = lane; h < F; h += 32) p += We[h] * ae[h];
  for (int m = 16; m >= 1; m >>= 1) p += __shfl_xor(p, m, 32);
  if (lane == 0) out[0] = p;
}

// ---------------- in-degree + edge_attr sum per dst ----------------
__global__ __launch_bounds__(256) void deg_kernel(const int* __restrict__ dst,
                                                  const float* __restrict__ ea,
                                                  float* __restrict__ deg,
                                                  float* __restrict__ sattr, int E) {
  const int e = blockIdx.x * 256 + threadIdx.x;
  if (e >= E) return;
  const int d = dst[e];
  atomicAdd(&deg[d], 1.0f);
  atomicAdd(&sattr[d], ea[e]);
}

__global__ __launch_bounds__(256) void la_kernel(const float* __restrict__ deg,
                                                 float* __restrict__ sattr, int N) {
  const int i = blockIdx.x * 256 + threadIdx.x;
  if (i >= N) return;
  sattr[i] = sattr[i] / fmaxf(deg[i], 1.0f);   // loop_attr, in place
}

// ---------------- self-loop logit + max-key init ----------------
__global__ __launch_bounds__(256) void prep_kernel(const float* __restrict__ a_src,
                                                   const float* __restrict__ a_dst,
                                                   const float* __restrict__ la,
                                                   const float* __restrict__ kedge,
                                                   float* __restrict__ slog,
                                                   unsigned* __restrict__ mkey, int N) {
  const int i = blockIdx.x * 256 + threadIdx.x;
  if (i >= N) return;
  const float l = lrelu(a_src[i] + a_dst[i] + la[i] * kedge[0]);
  slog[i] = l;
  mkey[i] = fkey(l);
}

// ---------------- edge logits + segment max (atomicMax on monotone key) ----------------
__global__ __launch_bounds__(256) void elogit_kernel(const int* __restrict__ src,
                                                     const int* __restrict__ dst,
                                                     const float* __restrict__ ea,
                                                     const float* __restrict__ a_src,
                                                     const float* __restrict__ a_dst,
                                                     const float* __restrict__ kedge,
                                                     float* __restrict__ logit,
                                                     unsigned* __restrict__ mkey, int E) {
  const int e = blockIdx.x * 256 + threadIdx.x;
  if (e >= E) return;
  const float l = lrelu(a_src[src[e]] + a_dst[dst[e]] + ea[e] * kedge[0]);
  logit[e] = l;
  atomicMax(&mkey[dst[e]], fkey(l));
}

// ---------------- wave-per-edge weighted scatter-add ----------------
__global__ __launch_bounds__(256) void eaggr_kernel(const int* __restrict__ src,
                                                    const int* __restrict__ dst,
                                                    const float* __restrict__ logit,
                                                    const unsigned* __restrict__ mkey,
                                                    const float* __restrict__ xl,
                                                    float* __restrict__ denom,
                                                    float* __restrict__ acc,
                                                    int E, int F) {
  const int e = blockIdx.x * 8 + (threadIdx.x >> 5);
  if (e >= E) return;
  const int lane = threadIdx.x & 31;
  const int s = src[e], d = dst[e];
  const float p = __expf(logit[e] - funkey(mkey[d]));
  if (lane == 0) atomicAdd(&denom[d], p);
  const float* xs = xl + (size_t)s * F;
  float* ad = acc + (size_t)d * F;
  for (int h = lane; h < F; h += 32) atomicAdd(&ad[h], p * xs[h]);
}

// ---------------- finalize: self-loop term, normalize, bias, PReLU ----------------
__global__ __launch_bounds__(256) void fin_kernel(const float* __restrict__ acc,
                                                  const float* __restrict__ xl,
                                                  const float* __restrict__ slog,
                                                  const unsigned* __restrict__ mkey,
                                                  const float* __restrict__ denom,
                                                  const float* __restrict__ bias,
                                                  const float* __restrict__ prelu_a,
                                                  float* __restrict__ out, int N, int F) {
  const int i = blockIdx.x * 8 + (threadIdx.x >> 5);
  if (i >= N) return;
  const int lane = threadIdx.x & 31;
  const float m = funkey(mkey[i]);
  const float ps = __expf(slog[i] - m);
  const float dn = denom[i] + ps;
  const float a = prelu_a[0];
  const float* ai = acc + (size_t)i * F;
  const float* xi = xl + (size_t)i * F;
  float* oi = out + (size_t)i * F;
  for (int h = lane; h < F; h += 32) {
    float v = (ai[h] + ps * xi[h]) / dn + bias[h];
    oi[h] = v > 0.0f ? v : a * v;
  }
}

extern "C" void kernel_launch(void* const* d_in, const int* in_sizes, int n_in,
                              void* d_out, int out_size, void* d_ws, size_t ws_size,
                              hipStream_t stream) {
  (void)n_in; (void)out_size; (void)ws_size;
  const float* x   = (const float*)d_in[0];
  const int*   ei  = (const int*)d_in[1];
  const float* ea  = (const float*)d_in[2];
  const float* W1  = (const float*)d_in[3];
  const float* as1 = (const float*)d_in[4];
  const float* ad1 = (const float*)d_in[5];
  const float* We1 = (const float*)d_in[6];
  const float* ae1 = (const float*)d_in[7];
  const float* b1  = (const float*)d_in[8];
  const float* W2  = (const float*)d_in[9];
  const float* as2 = (const float*)d_in[10];
  const float* ad2 = (const float*)d_in[11];
  const float* We2 = (const float*)d_in[12];
  const float* ae2 = (const float*)d_in[13];
  const float* b2  = (const float*)d_in[14];
  const float* pa  = (const float*)d_in[15];

  const int N = in_sizes[0] / 128;   // 100000
  const int E = in_sizes[2];         // 1600000
  const int* src = ei;
  const int* dst = ei + E;

  float* ws    = (float*)d_ws;
  float* xl    = ws;                         // N*128 (layer2 reuses first N*64)
  float* h1    = xl + (size_t)N * 128;       // N*128
  float* acc   = h1 + (size_t)N * 128;       // N*128
  float* logit = acc + (size_t)N * 128;      // E
  float* asrc  = logit + E;                  // N
  float* adst  = asrc + N;                   // N
  unsigned* mkey = (unsigned*)(adst + N);    // N
  float* denom = (float*)(mkey + N);         // N
  float* slog  = denom + N;                  // N
  float* deg   = slog + N;                   // N
  float* la    = deg + N;                    // N (sum attr -> loop_attr in place)
  float* ked   = la + N;                     // 2 scalars

  const int eb  = (E + 255) / 256;
  const int nb  = (N + 255) / 256;
  const int nwb = (N + 7) / 8;
  const int ewb = (E + 7) / 8;
  const int gtb = (((N + 15) / 16) + 7) / 8;

  // shared across both layers
  hipMemsetAsync(deg, 0, (size_t)N * 4, stream);
  hipMemsetAsync(la, 0, (size_t)N * 4, stream);
  deg_kernel<<<eb, 256, 0, stream>>>(dst, ea, deg, la, E);
  la_kernel<<<nb, 256, 0, stream>>>(deg, la, N);
  kedge_kernel<<<1, 32, 0, stream>>>(We1, ae1, 128, ked);
  kedge_kernel<<<1, 32, 0, stream>>>(We2, ae2, 64, ked + 1);

  // ---- layer 1 (128 -> 128) ----
  gemm_wmma<128, 128><<<gtb, 256, 0, stream>>>(x, W1, xl, N);
  dots_kernel<<<nwb, 256, 0, stream>>>(xl, as1, ad1, asrc, adst, N, 128);
  prep_kernel<<<nb, 256, 0, stream>>>(asrc, adst, la, ked, slog, mkey, N);
  hipMemsetAsync(denom, 0, (size_t)N * 4, stream);
  hipMemsetAsync(acc, 0, (size_t)N * 128 * 4, stream);
  elogit_kernel<<<eb, 256, 0, stream>>>(src, dst, ea, asrc, adst, ked, logit, mkey, E);
  eaggr_kernel<<<ewb, 256, 0, stream>>>(src, dst, logit, mkey, xl, denom, acc, E, 128);
  fin_kernel<<<nwb, 256, 0, stream>>>(acc, xl, slog, mkey, denom, b1, pa, h1, N, 128);

  // ---- layer 2 (128 -> 64) ----
  gemm_wmma<128, 64><<<gtb, 256, 0, stream>>>(h1, W2, xl, N);
  dots_kernel<<<nwb, 256, 0, stream>>>(xl, as2, ad2, asrc, adst, N, 64);
  prep_kernel<<<nb, 256, 0, stream>>>(asrc, adst, la, ked + 1, slog, mkey, N);
  hipMemsetAsync(denom, 0, (size_t)N * 4, stream);
  hipMemsetAsync(acc, 0, (size_t)N * 64 * 4, stream);
  elogit_kernel<<<eb, 256, 0, stream>>>(src, dst, ea, asrc, adst, ked + 1, logit, mkey, E);
  eaggr_kernel<<<ewb, 256, 0, stream>>>(src, dst, logit, mkey, xl, denom, acc, E, 64);
  fin_kernel<<<nwb, 256, 0, stream>>>(acc, xl, slog, mkey, denom, b2, pa, (float*)d_out, N, 64);
}